// DA3Loss_7687991459976
// MI455X (gfx1250) — compile-verified
//
#include <hip/hip_runtime.h>
#include <hip/hip_bf16.h>

// ---------------------------------------------------------------------------
// RANSAC depth alignment for gfx1250 (MI455X).
// All 100 RANSAC subsample fits are one tall-skinny f32 GEMM
//   sums[5x112] = V^T (N x 5, rows = valid*(1,p,g,p^2,p*g)) x S (N x 112 sel)
// executed with V_WMMA_F32_16X16X4_F32 (f32 in, f32 accumulate): the whole
// RANSAC sampling stage is a single streaming pass over ~26MB of inputs
// (~1.1us at 23.3 TB/s; everything after lives in the 192MB L2).
// Column 100 of S is all-ones -> full-data initial fit for free.
// This revision makes the A/B tile construction fully branch-free:
//  - per-lane one-hot coefficients replace the divergent quantity switch
//  - one hash per (element,row) supplies 7 selection bits (one per tile)
// ---------------------------------------------------------------------------

#define EPSV 1e-8f
#define ITERS 100
#define TILES 7            // 7 * 16 = 112 columns >= 101
#define COLS 112
#define NB 4096            // histogram bins for median

// workspace layout (word offsets into d_ws)
#define SUMS   0           // float[112*5]  sums[col*5 + r]
#define SARR   560         // float[100] per-iteration scale
#define BARR   660         // float[100] per-iteration shift
#define SCAL   760         // [0]=s0 [1]=b0 [2]=rmax [3]=thresh [4]=bs [5]=bb
                           // [6]=scale [7]=shift [8..12]=inlier sums
#define NVALID 776         // uint
#define HIST   1024        // uint[4096]
#define COUNTS 5200        // uint[100]

typedef float v2f __attribute__((ext_vector_type(2)));
typedef float v8f __attribute__((ext_vector_type(8)));

// one hash per (element, row 0..15) -> 7 independent selection bits (bit t for
// column 16*t + row)
__device__ __forceinline__ unsigned selhash(unsigned e, unsigned r) {
    unsigned h = (e * 2654435761u) ^ (r * 0x9E3779B9u);
    h ^= h >> 16; h *= 0x7FEB352Du;
    h ^= h >> 15; h *= 0x846CA68Bu;
    h ^= h >> 16;
    return h;
}

__device__ __forceinline__ void affine_fit(float n, float sp, float sg,
                                           float spp, float spg,
                                           float& s, float& b) {
    float det = spp * n - sp * sp;
    bool use_full = fabsf(det) >= EPSV;
    float det_safe = use_full ? det : 1.0f;
    float s_full = (spg * n - sp * sg) / det_safe;
    float b_full = (spp * sg - sp * spg) / det_safe;
    float s_deg = spg / fmaxf(spp, EPSV);
    s = use_full ? s_full : s_deg;
    b = use_full ? b_full : 0.0f;
}

// ---------------------------------------------------------------------------
__global__ void zero_ws(float* wsf, unsigned* wsu) {
    int t = threadIdx.x;
    for (int i = t; i < 800; i += 256) wsf[i] = 0.0f;          // sums + scalars
    for (int i = t; i < NB; i += 256) wsu[HIST + i] = 0u;
    for (int i = t; i < 128; i += 256) wsu[COUNTS + i] = 0u;
}

// ---------------------------------------------------------------------------
// Pass 1: WMMA GEMM — per-iteration (and full) weighted sums in one data pass.
__global__ __launch_bounds__(256) void ransac_gemm(const float* __restrict__ pred,
                                                   const float* __restrict__ gt,
                                                   const int* __restrict__ valid,
                                                   float* __restrict__ sums, int n) {
    __shared__ float lsum[COLS * 5];
    for (int i = threadIdx.x; i < COLS * 5; i += blockDim.x) lsum[i] = 0.0f;
    __syncthreads();

    const int lane = threadIdx.x & 31;
    const int wib = threadIdx.x >> 5;
    const int wpb = blockDim.x >> 5;
    const int gwave = blockIdx.x * wpb + wib;
    const int nwaves = gridDim.x * wpb;
    const int stride = nwaves * 32;

    v8f c[TILES];
    for (int t = 0; t < TILES; ++t)
        for (int j = 0; j < 8; ++j) c[t][j] = 0.0f;

    const int mrow = lane & 15;
    const int ka0 = (lane < 16) ? 0 : 2;   // A-matrix K slot for v0 (ISA 7.12.2)
    const int kb0 = (lane < 16) ? 0 : 1;   // B-matrix K slot for v0

    // branch-free one-hot quantity coefficients:
    // q(m) = w * (c1 + cp*p + cg*g + cpp*p^2 + cpg*p*g)
    const float c1  = (mrow == 0) ? 1.0f : 0.0f;
    const float cp  = (mrow == 1) ? 1.0f : 0.0f;
    const float cg  = (mrow == 2) ? 1.0f : 0.0f;
    const float cpp = (mrow == 3) ? 1.0f : 0.0f;
    const float cpg = (mrow == 4) ? 1.0f : 0.0f;
    // tile 6 column handling: cols 96..99 random, col 100 all-ones, 101+ zero
    const float t6rand = (mrow < 4) ? 1.0f : 0.0f;
    const float t6one  = (mrow == 4) ? 1.0f : 0.0f;

    for (int base = gwave * 32; base < n; base += stride) {
        int e = base + lane;
        float p = 0.0f, g = 0.0f, w = 0.0f;
        if (e < n) {
            p = pred[e];
            g = gt[e];
            w = (valid[e] != 0 && g > EPSV && p > EPSV) ? 1.0f : 0.0f;
        }
        // stream-ahead prefetch (global_prefetch_b8) for the next group
        __builtin_prefetch(pred + base + stride + lane, 0, 3);
        __builtin_prefetch(gt + base + stride + lane, 0, 3);

        #pragma unroll
        for (int s = 0; s < 8; ++s) {       // 8 K-steps of 4 elements = 32 elems
            int sl = 4 * s + ka0;
            float p0 = __shfl(p, sl),     g0 = __shfl(g, sl),     w0 = __shfl(w, sl);
            float p1 = __shfl(p, sl + 1), g1 = __shfl(g, sl + 1), w1 = __shfl(w, sl + 1);

            v2f a;
            a.x = w0 * fmaf(cpg, p0 * g0,
                      fmaf(cpp, p0 * p0, fmaf(cg, g0, fmaf(cp, p0, c1))));
            a.y = w1 * fmaf(cpg, p1 * g1,
                      fmaf(cpp, p1 * p1, fmaf(cg, g1, fmaf(cp, p1, c1))));

            unsigned eb0 = (unsigned)(base + 4 * s + kb0);
            unsigned h0 = selhash(eb0, (unsigned)mrow);
            unsigned h1 = selhash(eb0 + 2u, (unsigned)mrow);

            #pragma unroll
            for (int t = 0; t < TILES; ++t) {
                v2f b;
                float bx = (float)((h0 >> t) & 1u);
                float by = (float)((h1 >> t) & 1u);
                if (t == TILES - 1) {        // fold col-100 / zero columns
                    bx = fmaf(bx, t6rand, t6one);
                    by = fmaf(by, t6rand, t6one);
                }
                b.x = bx;
                b.y = by;
                c[t] = __builtin_amdgcn_wmma_f32_16x16x4_f32(
                    false, a, false, b, (short)0, c[t], false, false);
            }
        }
    }

    // rows 0..4 of each C tile live in c[0..4] of lanes 0..15 (M = vgpr index)
    if (lane < 16) {
        #pragma unroll
        for (int t = 0; t < TILES; ++t) {
            int col = 16 * t + lane;
            #pragma unroll
            for (int r = 0; r < 5; ++r)
                atomicAdd(&lsum[col * 5 + r], c[t][r]);
        }
    }
    __syncthreads();
    for (int i = threadIdx.x; i < COLS * 5; i += blockDim.x) {
        float v = lsum[i];
        if (v != 0.0f) atomicAdd(&sums[i], v);
    }
}

// ---------------------------------------------------------------------------
// Pass 2: solve 2x2 LSQ per column (100 subsamples + full fit).
__global__ void fit_kernel(float* wsf, unsigned* wsu) {
    int t = threadIdx.x;
    if (t <= ITERS) {
        const float* s5 = wsf + SUMS + t * 5;
        float nn = s5[0], sp = s5[1], sg = s5[2], spp = s5[3], spg = s5[4];
        float sc, sh;
        affine_fit(nn, sp, sg, spp, spg, sc, sh);
        if (t < ITERS) {
            wsf[SARR + t] = sc;
            wsf[BARR + t] = sh;
        } else {
            wsf[SCAL + 0] = sc;
            wsf[SCAL + 1] = sh;
            wsu[NVALID] = (unsigned)(nn + 0.5f);
            // residual bound: p,g in [0,1] -> |g - (s p + b)| <= 1 + |s| + |b|
            wsf[SCAL + 2] = fmaxf(1.0f + fabsf(sc) + fabsf(sh), 1e-6f);
        }
    }
}

// ---------------------------------------------------------------------------
// Pass 3: residual histogram (for the lower-median threshold).
__global__ __launch_bounds__(256) void hist_kernel(const float* __restrict__ pred,
                                                   const float* __restrict__ gt,
                                                   const int* __restrict__ valid,
                                                   const float* wsf, unsigned* wsu, int n) {
    __shared__ unsigned h[NB];
    for (int i = threadIdx.x; i < NB; i += blockDim.x) h[i] = 0u;
    __syncthreads();
    float s0 = wsf[SCAL + 0], b0 = wsf[SCAL + 1], rmax = wsf[SCAL + 2];
    float inv = (float)NB / rmax;
    for (int e = blockIdx.x * blockDim.x + threadIdx.x; e < n;
         e += gridDim.x * blockDim.x) {
        float p = pred[e], g = gt[e];
        if (valid[e] != 0 && g > EPSV && p > EPSV) {
            float r = fabsf(g - (s0 * p + b0));
            int bin = (int)(r * inv);
            if (bin > NB - 1) bin = NB - 1;
            atomicAdd(&h[bin], 1u);
        }
    }
    __syncthreads();
    for (int i = threadIdx.x; i < NB; i += blockDim.x)
        if (h[i]) atomicAdd(&wsu[HIST + i], h[i]);
}

__global__ void median_kernel(float* wsf, unsigned* wsu) {
    unsigned nv = wsu[NVALID];
    float rmax = wsf[SCAL + 2];
    float med = 0.0f;
    if (nv > 0u) {
        unsigned k = (nv - 1u) >> 1;
        unsigned cum = 0u;
        int bin = NB - 1;
        for (int i = 0; i < NB; ++i) {
            cum += wsu[HIST + i];
            if (cum > k) { bin = i; break; }
        }
        med = (bin + 0.5f) * rmax / (float)NB;
    }
    wsf[SCAL + 3] = 1.5f * med;
}

// ---------------------------------------------------------------------------
// Pass 4: inlier counts for all 100 models in one pass (wave32 ballot+popc).
__global__ __launch_bounds__(256) void count_kernel(const float* __restrict__ pred,
                                                    const float* __restrict__ gt,
                                                    const int* __restrict__ valid,
                                                    const float* wsf, unsigned* wsu, int n) {
    __shared__ float ss[ITERS], sb[ITERS];
    __shared__ unsigned lc[ITERS];
    for (int i = threadIdx.x; i < ITERS; i += blockDim.x) {
        ss[i] = wsf[SARR + i];
        sb[i] = wsf[BARR + i];
        lc[i] = 0u;
    }
    __syncthreads();
    float th = wsf[SCAL + 3];
    int lane = threadIdx.x & 31;
    for (int base = blockIdx.x * blockDim.x; base < n;
         base += gridDim.x * blockDim.x) {
        int e = base + threadIdx.x;
        float p = 0.0f, g = 0.0f;
        bool v = false;
        if (e < n) {
            p = pred[e];
            g = gt[e];
            v = (valid[e] != 0) && g > EPSV && p > EPSV;
        }
        for (int i = 0; i < ITERS; ++i) {
            bool in = v && (fabsf(g - (ss[i] * p + sb[i])) < th);
            unsigned long long m = __ballot(in);
            if (lane == 0 && m) atomicAdd(&lc[i], (unsigned)__popcll(m));
        }
    }
    __syncthreads();
    for (int i = threadIdx.x; i < ITERS; i += blockDim.x)
        if (lc[i]) atomicAdd(&wsu[COUNTS + i], lc[i]);
}

__global__ void best_kernel(float* wsf, unsigned* wsu) {
    unsigned best = 0u;
    float bs = 1.0f, bb = 0.0f;       // reference init: (0, 1.0, 0.0)
    for (int i = 0; i < ITERS; ++i) {
        unsigned c = wsu[COUNTS + i];
        if (c > best) {               // strict > keeps earliest max (scan order)
            best = c;
            bs = wsf[SARR + i];
            bb = wsf[BARR + i];
        }
    }
    wsf[SCAL + 4] = bs;
    wsf[SCAL + 5] = bb;
}

// ---------------------------------------------------------------------------
// Pass 5: weighted sums over inliers of the best model.
__global__ __launch_bounds__(256) void inlsum_kernel(const float* __restrict__ pred,
                                                     const float* __restrict__ gt,
                                                     const int* __restrict__ valid,
                                                     float* wsf, int n) {
    __shared__ float acc[5];
    if (threadIdx.x < 5) acc[threadIdx.x] = 0.0f;
    __syncthreads();
    float bs = wsf[SCAL + 4], bb = wsf[SCAL + 5], th = wsf[SCAL + 3];
    float a0 = 0, a1 = 0, a2 = 0, a3 = 0, a4 = 0;
    for (int e = blockIdx.x * blockDim.x + threadIdx.x; e < n;
         e += gridDim.x * blockDim.x) {
        float p = pred[e], g = gt[e];
        if (valid[e] != 0 && g > EPSV && p > EPSV &&
            fabsf(g - (bs * p + bb)) < th) {
            a0 += 1.0f; a1 += p; a2 += g; a3 += p * p; a4 += p * g;
        }
    }
    atomicAdd(&acc[0], a0);
    atomicAdd(&acc[1], a1);
    atomicAdd(&acc[2], a2);
    atomicAdd(&acc[3], a3);
    atomicAdd(&acc[4], a4);
    __syncthreads();
    if (threadIdx.x < 5 && acc[threadIdx.x] != 0.0f)
        atomicAdd(&wsf[SCAL + 8 + threadIdx.x], acc[threadIdx.x]);
}

__global__ void final_fit_kernel(float* wsf, float* out, int n) {
    float nn  = wsf[SCAL + 8],  sp  = wsf[SCAL + 9],  sg = wsf[SCAL + 10];
    float spp = wsf[SCAL + 11], spg = wsf[SCAL + 12];
    float sfit, bfit;
    affine_fit(nn, sp, sg, spp, spg, sfit, bfit);
    bool use_refit = nn > 10.5f;      // inlier count > 10
    float scale = use_refit ? sfit : wsf[SCAL + 4];
    float shift = use_refit ? bfit : wsf[SCAL + 5];
    scale = fminf(fmaxf(scale, 0.01f), 100.0f);
    wsf[SCAL + 6] = scale;
    wsf[SCAL + 7] = shift;
    out[n] = scale;                   // tuple: (aligned, scale, shift) flat
    out[n + 1] = shift;
}

__global__ __launch_bounds__(256) void apply_kernel(const float* __restrict__ pred,
                                                    const float* wsf,
                                                    float* __restrict__ out, int n) {
    float scale = wsf[SCAL + 6], shift = wsf[SCAL + 7];
    for (int e = blockIdx.x * blockDim.x + threadIdx.x; e < n;
         e += gridDim.x * blockDim.x)
        out[e] = fmaxf(scale * pred[e] + shift, 0.0f);
}

// ---------------------------------------------------------------------------
extern "C" void kernel_launch(void* const* d_in, const int* in_sizes, int n_in,
                              void* d_out, int out_size, void* d_ws, size_t ws_size,
                              hipStream_t stream) {
    const float* pred = (const float*)d_in[0];
    const float* gt   = (const float*)d_in[1];
    const int* valid  = (const int*)d_in[2];
    int n = in_sizes[0];
    float* out = (float*)d_out;
    float* wsf = (float*)d_ws;
    unsigned* wsu = (unsigned*)d_ws;

    zero_ws<<<1, 256, 0, stream>>>(wsf, wsu);
    ransac_gemm<<<512, 256, 0, stream>>>(pred, gt, valid, wsf + SUMS, n);
    fit_kernel<<<1, 128, 0, stream>>>(wsf, wsu);
    hist_kernel<<<512, 256, 0, stream>>>(pred, gt, valid, wsf, wsu, n);
    median_kernel<<<1, 1, 0, stream>>>(wsf, wsu);
    count_kernel<<<512, 256, 0, stream>>>(pred, gt, valid, wsf, wsu, n);
    best_kernel<<<1, 1, 0, stream>>>(wsf, wsu);
    inlsum_kernel<<<512, 256, 0, stream>>>(pred, gt, valid, wsf, n);
    final_fit_kernel<<<1, 1, 0, stream>>>(wsf, out, n);
    apply_kernel<<<1024, 256, 0, stream>>>(pred, wsf, out, n);
}